// Covariance_40492951667323
// MI455X (gfx1250) — compile-verified
//
#include <hip/hip_runtime.h>

typedef __attribute__((ext_vector_type(2))) float v2f;
typedef __attribute__((ext_vector_type(8))) float v8f;

#define B_      32
#define N_      8192
#define D_      128
#define KCHUNKS 16
#define CHUNK   (N_ / KCHUNKS)   // 512 rows per workgroup
#define KC      32               // rows staged in LDS per pipeline stage
#define STAGES  (CHUNK / KC)     // 16
#define SP      136              // padded LDS row stride (floats); 2*SP % 64 == 16 -> no half-wave bank alias
#define BUFSZ   (KC * SP)        // floats per LDS buffer

// ---------------------------------------------------------------------------
// Zero the Gram accumulator (d_out) and the per-batch column sums (d_ws).
// ---------------------------------------------------------------------------
__global__ void cov_zero_kernel(float* __restrict__ out, float* __restrict__ sums) {
    int i = blockIdx.x * blockDim.x + threadIdx.x;
    if (i < B_ * D_ * D_) out[i] = 0.0f;
    if (i < B_ * D_)      sums[i] = 0.0f;
}

// ---------------------------------------------------------------------------
// Gram + column-sum kernel. One workgroup per (batch, K-chunk).
// 4 waves; wave w computes output region d in [ (w>>1)*64, +64 ), e in [ (w&1)*64, +64 )
// as a 4x4 grid of 16x16 f32 WMMA accumulator tiles.
// Staging uses CDNA5 async global->LDS DMA (ASYNCcnt) with double buffering.
// ---------------------------------------------------------------------------
__global__ __launch_bounds__(128) void cov_gram_kernel(const float* __restrict__ x,
                                                       float* __restrict__ gram,
                                                       float* __restrict__ sums) {
    __shared__ alignas(16) float lds[2 * BUFSZ];

    const int b     = blockIdx.x / KCHUNKS;
    const int chunk = blockIdx.x % KCHUNKS;
    const int t     = threadIdx.x;
    const int lane  = t & 31;
    const int wid   = t >> 5;
    const int half  = lane >> 4;   // which 16-lane half of the wave
    const int lc    = lane & 15;   // lane within half
    const int dbase = (wid >> 1) * 64;
    const int ebase = (wid & 1)  * 64;

    const float* xb = x + (size_t)b * N_ * D_ + (size_t)chunk * CHUNK * D_;

    // Byte address of LDS buffer start (low 32 bits of the flat address are the
    // LDS byte offset per the gfx1250 aperture rules).
    const unsigned ldsBase = (unsigned)(size_t)(&lds[0]);

    // Issue one stage's async DMA: KC*D_ floats = KC*D_/4 float4 lines,
    // 8 per thread. Per-lane LDS dest address + per-lane global offset,
    // uniform SGPR base (GVS addressing).
    auto issue_stage = [&](int s, int buf) {
        const char* gbase = (const char*)(xb + (size_t)s * KC * D_);
        const unsigned lbase = ldsBase + (unsigned)(buf * BUFSZ * 4);
#pragma unroll
        for (int i = 0; i < (KC * D_ / 4) / 128; ++i) {   // 8 iterations
            int      v     = i * 128 + t;                 // float4 index within stage
            unsigned row   = (unsigned)(v >> 5);          // D_/4 = 32 float4 per row
            unsigned c4    = (unsigned)((v & 31) * 4);    // float col
            unsigned laddr = lbase + (row * SP + c4) * 4; // LDS byte address
            unsigned goff  = (unsigned)v * 16;            // global byte offset
            asm volatile("global_load_async_to_lds_b128 %0, %1, %2"
                         :
                         : "v"(laddr), "v"(goff), "s"(gbase)
                         : "memory");
        }
    };

    v8f acc[4][4];
    const v8f vzero = {0.f, 0.f, 0.f, 0.f, 0.f, 0.f, 0.f, 0.f};
#pragma unroll
    for (int i = 0; i < 4; ++i)
#pragma unroll
        for (int j = 0; j < 4; ++j) acc[i][j] = vzero;

    float csum = 0.0f;  // thread t accumulates column t's sum (D_ == blockDim.x)

    issue_stage(0, 0);

    for (int s = 0; s < STAGES; ++s) {
        // Wait for this wave's outstanding async DMA (stage s), then make all
        // waves' DMA results visible to the whole workgroup.
        asm volatile("s_wait_asynccnt 0x0" ::: "memory");
        __syncthreads();

        // Prefetch next stage into the other buffer while we compute.
        if (s + 1 < STAGES) issue_stage(s + 1, (s + 1) & 1);

        const float* buf = &lds[(s & 1) * BUFSZ];

        // ---- per-column partial sums (free ride on staged data) ----
#pragma unroll 8
        for (int r = 0; r < KC; ++r) csum += buf[r * SP + t];

        // ---- WMMA sweep over this stage, K in steps of 4 ----
        for (int k0 = 0; k0 < KC; k0 += 4) {
            // Fragment layout (ISA 16x4 f32 A-matrix):
            //   VGPR0: lanes 0-15 -> K=k0,   lanes 16-31 -> K=k0+2
            //   VGPR1: lanes 0-15 -> K=k0+1, lanes 16-31 -> K=k0+3
            float a0[4], a1[4], b0[4], b1[4];
            const int rb = (k0 + 2 * half) * SP + lc;
#pragma unroll
            for (int j = 0; j < 4; ++j) {
                int oa = rb + dbase + j * 16;
                a0[j] = buf[oa];
                a1[j] = buf[oa + SP];
                int ob = rb + ebase + j * 16;
                b0[j] = buf[ob];
                b1[j] = buf[ob + SP];
            }
#pragma unroll
            for (int ti = 0; ti < 4; ++ti) {
                v2f A = {a0[ti], a1[ti]};
#pragma unroll
                for (int tj = 0; tj < 4; ++tj) {
                    v2f Bf = {b0[tj], b1[tj]};
                    acc[ti][tj] = __builtin_amdgcn_wmma_f32_16x16x4_f32(
                        /*neg_a=*/false, A, /*neg_b=*/false, Bf,
                        /*c_mod=*/(short)0, acc[ti][tj],
                        /*reuse_a=*/false, /*reuse_b=*/false);
                }
            }
        }
        __syncthreads();
    }

    // ---- reduce partial Gram into global accumulator ----
    // C/D tile layout: lane L, VGPR r -> element (M = 8*half + r, N = lc)
    float* gb = gram + (size_t)b * D_ * D_;
#pragma unroll
    for (int ti = 0; ti < 4; ++ti)
#pragma unroll
        for (int tj = 0; tj < 4; ++tj)
#pragma unroll
            for (int r = 0; r < 8; ++r) {
                int d = dbase + ti * 16 + half * 8 + r;
                int e = ebase + tj * 16 + lc;
                atomicAdd(&gb[d * D_ + e], acc[ti][tj][r]);
            }

    atomicAdd(&sums[b * D_ + t], csum);
}

// ---------------------------------------------------------------------------
// cov = G/N - (S_d/N)(S_e/N), in place on d_out.
// ---------------------------------------------------------------------------
__global__ void cov_finalize_kernel(float* __restrict__ out, const float* __restrict__ sums) {
    int i = blockIdx.x * blockDim.x + threadIdx.x;
    if (i >= B_ * D_ * D_) return;
    const float invN = 1.0f / (float)N_;
    int b = i / (D_ * D_);
    int r = i % (D_ * D_);
    int d = r / D_;
    int e = r % D_;
    float md = sums[b * D_ + d] * invN;
    float me = sums[b * D_ + e] * invN;
    out[i] = out[i] * invN - md * me;
}

extern "C" void kernel_launch(void* const* d_in, const int* in_sizes, int n_in,
                              void* d_out, int out_size, void* d_ws, size_t ws_size,
                              hipStream_t stream) {
    const float* x   = (const float*)d_in[0];
    float* out       = (float*)d_out;
    float* sums      = (float*)d_ws;        // 32*128 f32 = 16 KB of scratch

    // 1) zero accumulators
    {
        int n = B_ * D_ * D_;
        int blocks = (n + 255) / 256;
        cov_zero_kernel<<<blocks, 256, 0, stream>>>(out, sums);
    }
    // 2) WMMA Gram + column sums (async-DMA double-buffered)
    {
        dim3 grid(B_ * KCHUNKS);
        cov_gram_kernel<<<grid, 128, 0, stream>>>(x, out, sums);
    }
    // 3) finalize: cov = G/N - mu mu^T
    {
        int n = B_ * D_ * D_;
        int blocks = (n + 255) / 256;
        cov_finalize_kernel<<<blocks, 256, 0, stream>>>(out, sums);
    }
}